// DySkillHGNN_11055245820283
// MI455X (gfx1250) — compile-verified
//
#include <hip/hip_runtime.h>
#include <math.h>

#define N_NODES 30000
#define DIM     128
#define N_T     6
#define N_EDGES 480000
#define NSEL    1024
#define NEG_SLOPE 0.2f

typedef __attribute__((ext_vector_type(2))) float v2f;
typedef __attribute__((ext_vector_type(8))) float v8f;

__device__ __forceinline__ float leaky(float x) {
  return x > 0.f ? x : NEG_SLOPE * x;
}
// Monotonic float<->uint mapping so segment-max can use atomicMax(u32).
__device__ __forceinline__ unsigned f2ord(float x) {
  unsigned u = __float_as_uint(x);
  return (u & 0x80000000u) ? ~u : (u | 0x80000000u);
}
__device__ __forceinline__ float ord2f(unsigned o) {
  return __uint_as_float((o & 0x80000000u) ? (o & 0x7fffffffu) : ~o);
}

// ---------------- slot compaction (dst in s only) ----------------
__global__ void k_init_slot(int* slot, int* nslots) {
  int i = blockIdx.x * blockDim.x + threadIdx.x;
  if (i < N_NODES) slot[i] = -1;
  if (i == 0) *nslots = 0;
}
__global__ void k_mark(const int* __restrict__ s, int* slot) {
  int i = blockIdx.x * blockDim.x + threadIdx.x;
  if (i < NSEL) atomicCAS(&slot[s[i]], -1, -2);
}
__global__ void k_compact(int* slot, int* node_of, int* nslots) {
  int i = blockIdx.x * blockDim.x + threadIdx.x;
  if (i < N_NODES && slot[i] == -2) {
    int id = atomicAdd(nslots, 1);
    slot[i] = id;
    node_of[id] = i;
  }
}

// ---------------- H = X @ W for 4 weight matrices, f32 WMMA ----------------
// W staged in LDS (64 KB). One wave computes a full 16x128 row block:
// per k-step one A fragment (global b64) feeds 8 back-to-back WMMAs whose B
// fragments come from LDS. 30000 = 1875*16 row tiles exactly; 128 = 8*16 cols.
__global__ void __launch_bounds__(256)
k_gemm4(const float* __restrict__ X,
        const float* __restrict__ W0, const float* __restrict__ W1,
        const float* __restrict__ W2, const float* __restrict__ W3,
        float* __restrict__ H0, float* __restrict__ H1,
        float* __restrict__ H2, float* __restrict__ H3) {
  __shared__ float Wlds[DIM * DIM];   // 64 KB
  const float* W; float* H;
  switch (blockIdx.y) {
    case 0:  W = W0; H = H0; break;
    case 1:  W = W1; H = H1; break;
    case 2:  W = W2; H = H2; break;
    default: W = W3; H = H3; break;
  }
  // cooperative stage of W into LDS (float4, fully coalesced)
  {
    const float4* Wg = (const float4*)W;
    float4* Wl = (float4*)Wlds;
    for (int i = threadIdx.x; i < DIM * DIM / 4; i += blockDim.x) Wl[i] = Wg[i];
  }
  __syncthreads();

  int mt = blockIdx.x * (blockDim.x >> 5) + (threadIdx.x >> 5);
  if (mt >= N_NODES / 16) return;
  int lane = threadIdx.x & 31;
  int r    = lane & 15;
  int hi   = lane >> 4;          // lanes 16-31 carry K+2 / M+8 halves
  int row  = mt * 16 + r;
  const float* xrow = X + (size_t)row * DIM;

  v8f c[8];
  v8f z = {};
#pragma unroll
  for (int n = 0; n < 8; ++n) c[n] = z;

  for (int kb = 0; kb < DIM; kb += 4) {
    int k0 = kb + (hi << 1);
    v2f a; a.x = xrow[k0]; a.y = xrow[k0 + 1];
#pragma unroll
    for (int n = 0; n < 8; ++n) {
      int col = n * 16 + r;
      v2f b; b.x = Wlds[k0 * DIM + col]; b.y = Wlds[(k0 + 1) * DIM + col];
      c[n] = __builtin_amdgcn_wmma_f32_16x16x4_f32(false, a, false, b,
                                                   (short)0, c[n], false, false);
    }
  }
  int rbase = mt * 16 + hi * 8;
#pragma unroll
  for (int n = 0; n < 8; ++n)
#pragma unroll
    for (int v = 0; v < 8; ++v)
      H[(size_t)(rbase + v) * DIM + n * 16 + r] = c[n][v];
}

// ---------------- per-node attention scalars: h . att ----------------
__global__ void k_att(const float* __restrict__ hP, const float* __restrict__ hC,
                      const float* __restrict__ asp, const float* __restrict__ adp,
                      const float* __restrict__ asc, const float* __restrict__ adc,
                      float* aSP, float* aDP, float* aSC, float* aDC) {
  int wave = blockIdx.x * (blockDim.x >> 5) + (threadIdx.x >> 5);
  int lane = threadIdx.x & 31;
  if (wave >= N_NODES) return;
  float4 hp = ((const float4*)(hP + (size_t)wave * DIM))[lane];
  float4 hc = ((const float4*)(hC + (size_t)wave * DIM))[lane];
  float4 a1 = ((const float4*)asp)[lane];
  float4 a2 = ((const float4*)adp)[lane];
  float4 a3 = ((const float4*)asc)[lane];
  float4 a4 = ((const float4*)adc)[lane];
  float sp = hp.x*a1.x + hp.y*a1.y + hp.z*a1.z + hp.w*a1.w;
  float dp = hp.x*a2.x + hp.y*a2.y + hp.z*a2.z + hp.w*a2.w;
  float sc = hc.x*a3.x + hc.y*a3.y + hc.z*a3.z + hc.w*a3.w;
  float dc = hc.x*a4.x + hc.y*a4.y + hc.z*a4.z + hc.w*a4.w;
  for (int off = 16; off > 0; off >>= 1) {
    sp += __shfl_xor(sp, off);
    dp += __shfl_xor(dp, off);
    sc += __shfl_xor(sc, off);
    dc += __shfl_xor(dc, off);
  }
  if (lane == 0) { aSP[wave] = sp; aDP[wave] = dp; aSC[wave] = sc; aDC[wave] = dc; }
}

// ---------------- per-timestep scratch init ----------------
__global__ void k_init_t(float* accP, float* accC, float* accL,
                         unsigned* mP, unsigned* mC,
                         float* dP, float* dC, float* cnt) {
  int i = blockIdx.x * blockDim.x + threadIdx.x;
  if (i < NSEL * DIM) { accP[i] = 0.f; accC[i] = 0.f; accL[i] = 0.f; }
  if (i < NSEL) { mP[i] = 0u; mC[i] = 0u; dP[i] = 0.f; dC[i] = 0.f; cnt[i] = 0.f; }
}

// ---------------- GAT pass 1: segment max over selected dst ----------------
__global__ void k_gat_max(const int* __restrict__ src, const int* __restrict__ dst,
                          const float* __restrict__ aS, const float* __restrict__ aD,
                          const int* __restrict__ slot, const int* __restrict__ node_of,
                          const int* __restrict__ nslots, unsigned* mI) {
  int i = blockIdx.x * blockDim.x + threadIdx.x;
  int s_, d_, sl;
  if (i < N_EDGES) {
    d_ = dst[i]; sl = slot[d_];
    if (sl < 0) return;
    s_ = src[i];
  } else {
    sl = i - N_EDGES;
    if (sl >= *nslots) return;
    s_ = d_ = node_of[sl];
  }
  atomicMax(&mI[sl], f2ord(leaky(aS[s_] + aD[d_])));
}

// ---------------- GAT pass 2 (fused): unnormalized exp-weighted aggregation
// plus denominator accumulation; one wave per edge / self-loop. ----------------
__global__ void k_gat_agg(const int* __restrict__ src, const int* __restrict__ dst,
                          const float* __restrict__ aS, const float* __restrict__ aD,
                          const int* __restrict__ slot, const int* __restrict__ node_of,
                          const int* __restrict__ nslots,
                          const unsigned* __restrict__ mI,
                          float* den, float* acc,
                          const float* __restrict__ h) {
  int wave = blockIdx.x * (blockDim.x >> 5) + (threadIdx.x >> 5);
  int lane = threadIdx.x & 31;
  int s_, d_, sl;
  if (wave < N_EDGES) {
    d_ = dst[wave]; sl = slot[d_];
    if (sl < 0) return;
    s_ = src[wave];
  } else {
    sl = wave - N_EDGES;
    if (sl >= *nslots) return;
    s_ = d_ = node_of[sl];
  }
  float e  = leaky(aS[s_] + aD[d_]);
  float ex = __expf(e - ord2f(mI[sl]));
  if (lane == 0) atomicAdd(&den[sl], ex);
  float4 hv = ((const float4*)(h + (size_t)s_ * DIM))[lane];
  float* a = acc + (size_t)sl * DIM + lane * 4;
  atomicAdd(a + 0, ex * hv.x);
  atomicAdd(a + 1, ex * hv.y);
  atomicAdd(a + 2, ex * hv.z);
  atomicAdd(a + 3, ex * hv.w);
}

// ---------------- SAGE: sum hL[src] into slots + indegree count ----------------
__global__ void k_sage(const int* __restrict__ src, const int* __restrict__ dst,
                       const int* __restrict__ slot,
                       const float* __restrict__ hL, float* accL, float* cnt) {
  int wave = blockIdx.x * (blockDim.x >> 5) + (threadIdx.x >> 5);
  int lane = threadIdx.x & 31;
  if (wave >= N_EDGES) return;
  int d_ = dst[wave]; int sl = slot[d_];
  if (sl < 0) return;
  int s_ = src[wave];
  if (lane == 0) atomicAdd(&cnt[sl], 1.f);
  float4 hv = ((const float4*)(hL + (size_t)s_ * DIM))[lane];
  float* a = accL + (size_t)sl * DIM + lane * 4;
  atomicAdd(a + 0, hv.x);
  atomicAdd(a + 1, hv.y);
  atomicAdd(a + 2, hv.z);
  atomicAdd(a + 3, hv.w);
}

// ---------------- combine: (GATp + GATc + SAGE)/3 gathered at s ----------------
__global__ void k_combine(const int* __restrict__ s, const int* __restrict__ slot,
                          const float* __restrict__ accP, const float* __restrict__ accC,
                          const float* __restrict__ accL,
                          const float* __restrict__ dP, const float* __restrict__ dC,
                          const float* __restrict__ cnt,
                          const float* __restrict__ hR,
                          const float* __restrict__ bP, const float* __restrict__ bC,
                          const float* __restrict__ bL, const float* __restrict__ bR,
                          float* __restrict__ out) {
  int i = blockIdx.x;    // 0..1023
  int d = threadIdx.x;   // 0..127
  int node = s[i];
  int sl = slot[node];
  float mean = accL[(size_t)sl * DIM + d] / fmaxf(cnt[sl], 1.f);
  float sage = mean + bL[d] + hR[(size_t)node * DIM + d] + bR[d];
  float gp = accP[(size_t)sl * DIM + d] / dP[sl] + bP[d];
  float gc = accC[(size_t)sl * DIM + d] / dC[sl] + bC[d];
  out[(size_t)i * DIM + d] = (gp + gc + sage) * (1.f / 3.f);
}

extern "C" void kernel_launch(void* const* d_in, const int* in_sizes, int n_in,
                              void* d_out, int out_size, void* d_ws, size_t ws_size,
                              hipStream_t stream) {
  const int*   s   = (const int*)d_in[0];
  const int*   ei  = (const int*)d_in[3];   // [6,3,2,480000]
  const float* X   = (const float*)d_in[4];
  const float* Wp  = (const float*)d_in[5];
  const float* asp = (const float*)d_in[6];
  const float* adp = (const float*)d_in[7];
  const float* bp  = (const float*)d_in[8];
  const float* Wc  = (const float*)d_in[9];
  const float* asc = (const float*)d_in[10];
  const float* adc = (const float*)d_in[11];
  const float* bc  = (const float*)d_in[12];
  const float* Wl  = (const float*)d_in[13];
  const float* bl  = (const float*)d_in[14];
  const float* Wr  = (const float*)d_in[15];
  const float* br  = (const float*)d_in[16];
  float* out = (float*)d_out;

  // workspace carve-out (~63.5 MB)
  char* w = (char*)d_ws;
  auto alloc = [&](size_t bytes) -> char* {
    char* p = w; w += (bytes + 255) & ~(size_t)255; return p;
  };
  float*    hP   = (float*)alloc((size_t)N_NODES * DIM * 4);
  float*    hC   = (float*)alloc((size_t)N_NODES * DIM * 4);
  float*    hL   = (float*)alloc((size_t)N_NODES * DIM * 4);
  float*    hR   = (float*)alloc((size_t)N_NODES * DIM * 4);
  float*    aSP  = (float*)alloc((size_t)N_NODES * 4);
  float*    aDP  = (float*)alloc((size_t)N_NODES * 4);
  float*    aSC  = (float*)alloc((size_t)N_NODES * 4);
  float*    aDC  = (float*)alloc((size_t)N_NODES * 4);
  int*      slot = (int*)alloc((size_t)N_NODES * 4);
  int*      nof  = (int*)alloc((size_t)NSEL * 4);
  int*      nsl  = (int*)alloc(256);
  unsigned* mP   = (unsigned*)alloc((size_t)NSEL * 4);
  unsigned* mC   = (unsigned*)alloc((size_t)NSEL * 4);
  float*    dP   = (float*)alloc((size_t)NSEL * 4);
  float*    dC   = (float*)alloc((size_t)NSEL * 4);
  float*    cnt  = (float*)alloc((size_t)NSEL * 4);
  float*    accP = (float*)alloc((size_t)NSEL * DIM * 4);
  float*    accC = (float*)alloc((size_t)NSEL * DIM * 4);
  float*    accL = (float*)alloc((size_t)NSEL * DIM * 4);

  // 1) slot compaction for selected dst nodes
  k_init_slot<<<(N_NODES + 255) / 256, 256, 0, stream>>>(slot, nsl);
  k_mark<<<(NSEL + 255) / 256, 256, 0, stream>>>(s, slot);
  k_compact<<<(N_NODES + 255) / 256, 256, 0, stream>>>(slot, nof, nsl);

  // 2) timestep-invariant GEMMs (WMMA f32, LDS-staged W): hP, hC, hL, hR
  //    1875 row tiles, 8 waves (= 8 row tiles) per block
  k_gemm4<<<dim3((1875 + 7) / 8, 4), 256, 0, stream>>>(X, Wp, Wc, Wl, Wr,
                                                       hP, hC, hL, hR);

  // 3) attention scalars
  k_att<<<(N_NODES + 7) / 8, 256, 0, stream>>>(hP, hC, asp, adp, asc, adc,
                                               aSP, aDP, aSC, aDC);

  const int edgeDom  = N_EDGES + NSEL;          // edges + self-loop slots
  const int gridScal = (edgeDom + 255) / 256;   // thread-per-edge kernels
  const int gridWave = (edgeDom + 7) / 8;       // wave-per-edge kernels
  const int gridSage = (N_EDGES + 7) / 8;

  for (int t = 0; t < N_T; ++t) {
    const int* srcP = ei + ((size_t)(t * 3 + 0) * 2 + 0) * N_EDGES;
    const int* dstP = ei + ((size_t)(t * 3 + 0) * 2 + 1) * N_EDGES;
    const int* srcC = ei + ((size_t)(t * 3 + 1) * 2 + 0) * N_EDGES;
    const int* dstC = ei + ((size_t)(t * 3 + 1) * 2 + 1) * N_EDGES;
    const int* srcL = ei + ((size_t)(t * 3 + 2) * 2 + 0) * N_EDGES;
    const int* dstL = ei + ((size_t)(t * 3 + 2) * 2 + 1) * N_EDGES;

    k_init_t<<<(NSEL * DIM + 255) / 256, 256, 0, stream>>>(accP, accC, accL,
                                                           mP, mC, dP, dC, cnt);
    // GAT "p": max pass, then fused exp-agg + denominator pass
    k_gat_max<<<gridScal, 256, 0, stream>>>(srcP, dstP, aSP, aDP, slot, nof, nsl, mP);
    k_gat_agg<<<gridWave, 256, 0, stream>>>(srcP, dstP, aSP, aDP, slot, nof, nsl,
                                            mP, dP, accP, hP);
    // GAT "c"
    k_gat_max<<<gridScal, 256, 0, stream>>>(srcC, dstC, aSC, aDC, slot, nof, nsl, mC);
    k_gat_agg<<<gridWave, 256, 0, stream>>>(srcC, dstC, aSC, aDC, slot, nof, nsl,
                                            mC, dC, accC, hC);
    // SAGE
    k_sage<<<gridSage, 256, 0, stream>>>(srcL, dstL, slot, hL, accL, cnt);

    // combine (normalize by denominators here) + gather to output
    k_combine<<<NSEL, DIM, 0, stream>>>(s, slot, accP, accC, accL, dP, dC, cnt, hR,
                                        bp, bc, bl, br,
                                        out + (size_t)t * NSEL * DIM);
  }
}